// _FixedIterMaxBP_14516989461226
// MI455X (gfx1250) — compile-verified
//
#include <hip/hip_runtime.h>
#include <math.h>
#include <stdint.h>

// Problem constants (from reference)
#define NVAR 50000
#define NEDGE 200000
#define NS 8
#define TOTAL_STEPS 7   // 1 initial bp_step + MAXITER(6) scan steps
#define DAMP 0.5f

typedef float v2f __attribute__((ext_vector_type(2)));
typedef float v8f __attribute__((ext_vector_type(8)));

#define EPB 160        // edges per block in edge kernel (200000 % 160 == 0)
#define PW_STRIDE 66   // padded LDS stride (floats) per edge: kills bank conflicts

// ---------------------------------------------------------------------------
// vb = unary  (then scatter_kernel accumulates messages on top)
__global__ __launch_bounds__(256) void init_vb_kernel(const float* __restrict__ unary,
                                                      float* __restrict__ vb) {
  int i = blockIdx.x * blockDim.x + threadIdx.x;
  if (i < NVAR * NS) vb[i] = unary[i];
}

// vb[var, s] += msgs[dir, e, s]   (one thread per (dir,e,s); flat index == tid)
__global__ __launch_bounds__(256) void scatter_kernel(const float* __restrict__ msgs,
                                                      const int* __restrict__ eidx,
                                                      float* __restrict__ vb) {
  int tid = blockIdx.x * blockDim.x + threadIdx.x;
  if (tid >= 2 * NEDGE * NS) return;
  int dir = tid / (NEDGE * NS);
  int rem = tid - dir * (NEDGE * NS);
  int e = rem >> 3;
  int s = rem & 7;
  int var = eidx[dir * NEDGE + e];
  atomicAdd(&vb[var * NS + s], msgs[tid]);
}

// ---------------------------------------------------------------------------
// Per-edge max-plus update. Pairwise tile staged into LDS via CDNA5 async copy.
__global__ __launch_bounds__(EPB) void edge_kernel(
    const float* __restrict__ msgs_in, float* __restrict__ msgs_out,
    const float* __restrict__ vb, const float* __restrict__ pairwise,
    const int* __restrict__ eidx, float* __restrict__ fb_out,
    const unsigned char* __restrict__ fmask) {
  __shared__ float pw[EPB * PW_STRIDE];
  const int t = threadIdx.x;
  const long long e0 = (long long)blockIdx.x * EPB;

  // ---- CDNA5 async global->LDS staging (ASYNCcnt), coalesced 16B chunks ----
  {
    unsigned lds_base = (unsigned)(uintptr_t)(void*)pw;
    const float* gsrc = pairwise + e0 * 64;
#pragma unroll
    for (int k = 0; k < 16; ++k) {
      unsigned c = (unsigned)t + (unsigned)k * EPB;  // chunk id in [0, EPB*16)
      unsigned goff = c * 16u;                       // global byte offset
      unsigned loff = lds_base + (c >> 4) * (PW_STRIDE * 4u) + (c & 15u) * 16u;
      asm volatile("global_load_async_to_lds_b128 %0, %1, %2"
                   :: "v"(loff), "v"(goff), "s"(gsrc) : "memory");
    }
    asm volatile("s_wait_asynccnt 0x0" ::: "memory");
  }
  __syncthreads();

  const long long e = e0 + t;
  const int ue = eidx[e];
  const int ve = eidx[NEDGE + e];

  float m0[8], m1[8], nu[8], nv[8];
  const float* mi0 = msgs_in + e * 8;
  const float* mi1 = msgs_in + (long long)NEDGE * 8 + e * 8;
  const float* bu = vb + (long long)ue * 8;
  const float* bv = vb + (long long)ve * 8;
#pragma unroll
  for (int s = 0; s < 8; ++s) {
    m0[s] = mi0[s]; m1[s] = mi1[s];
    nu[s] = bu[s] - m0[s];   // var->factor message toward this factor (dir u)
    nv[s] = bv[s] - m1[s];   // (dir v)
  }

  float nm0[8], nm1[8];
#pragma unroll
  for (int j = 0; j < 8; ++j) nm1[j] = -__builtin_inff();

  const float* rowbase = pw + t * PW_STRIDE;
#pragma unroll
  for (int i = 0; i < 8; ++i) {
    float r[8];
#pragma unroll
    for (int j = 0; j < 8; ++j) r[j] = rowbase[i * 8 + j];
    float best = -__builtin_inff();
#pragma unroll
    for (int j = 0; j < 8; ++j) {
      best = fmaxf(best, r[j] + nv[j]);          // nm0[i] = max_j P[i,j]+n_v[j]
      nm1[j] = fmaxf(nm1[j], r[j] + nu[i]);      // nm1[j] = max_i P[i,j]+n_u[i]
    }
    nm0[i] = best;
    if (fb_out) {  // last step: raw (mask-applied) factor beliefs, calibrated later
#pragma unroll
      for (int j = 0; j < 8; ++j) {
        float val = r[j] + nu[i] + nv[j];
        if (fmask[e * 64 + i * 8 + j]) val = -__builtin_inff();
        fb_out[e * 64 + i * 8 + j] = val;
      }
    }
  }

  // normalize (per-message max -> 0) then damp
  float mx0 = -__builtin_inff(), mx1 = -__builtin_inff();
#pragma unroll
  for (int s = 0; s < 8; ++s) { mx0 = fmaxf(mx0, nm0[s]); mx1 = fmaxf(mx1, nm1[s]); }
  float* mo0 = msgs_out + e * 8;
  float* mo1 = msgs_out + (long long)NEDGE * 8 + e * 8;
#pragma unroll
  for (int s = 0; s < 8; ++s) {
    mo0[s] = DAMP * m0[s] + (1.0f - DAMP) * (nm0[s] - mx0);
    mo1[s] = DAMP * m1[s] + (1.0f - DAMP) * (nm1[s] - mx1);
  }
}

// ---------------------------------------------------------------------------
// Masked softmax over the 8 states of each variable belief.
__global__ __launch_bounds__(256) void calibrate_vb_kernel(const float* __restrict__ vb,
                                                           const unsigned char* __restrict__ vmask,
                                                           float* __restrict__ out) {
  int v = blockIdx.x * blockDim.x + threadIdx.x;
  if (v >= NVAR) return;
  float b[8];
  float m = -__builtin_inff();
#pragma unroll
  for (int s = 0; s < 8; ++s) {
    b[s] = vmask[v * 8 + s] ? -__builtin_inff() : vb[v * 8 + s];
    m = fmaxf(m, b[s]);
  }
  float ex[8], sum = 0.0f;
#pragma unroll
  for (int s = 0; s < 8; ++s) { ex[s] = expf(b[s] - m); sum += ex[s]; }
#pragma unroll
  for (int s = 0; s < 8; ++s) out[v * 8 + s] = ex[s] / sum;
}

// ---------------------------------------------------------------------------
// Masked softmax over 64 factor-belief entries per edge, 16 edges per wave.
// The 64-wide exp-sum is computed as D = sum_c A_c x Ones with
// V_WMMA_F32_16X16X4_F32 (exact f32 matrix-engine reduction).
// A layout (32-bit A 16x4): lanes 0-15 give K=0,1; lanes 16-31 give K=2,3.
__global__ __launch_bounds__(128) void calibrate_fb_kernel(float* __restrict__ fb) {
  __shared__ float red[4 * 256];  // 4 waves x 32 lanes x 8 accum VGPRs
  const int lane = threadIdx.x & 31;
  const int wv = threadIdx.x >> 5;
  const int eloc = lane & 15;   // edge within wave's 16-edge group (matrix row M)
  const int half = lane >> 4;   // 0 -> K%4 in {0,1}, 1 -> K%4 in {2,3}
  const long long ebase = (long long)blockIdx.x * 64 + (long long)wv * 16;
  float* p = fb + (ebase + eloc) * 64 + half * 2;

  // pass 1: load 32 of the edge's 64 values per lane, track local max
  float x[32];
  float mloc = -__builtin_inff();
#pragma unroll
  for (int c = 0; c < 16; ++c) {
    v2f v = *(const v2f*)(p + c * 4);
    x[2 * c] = v.x;
    x[2 * c + 1] = v.y;
    mloc = fmaxf(mloc, fmaxf(v.x, v.y));
  }
  // wave32: lane l and l+16 jointly own one edge
  float m = fmaxf(mloc, __shfl_xor(mloc, 16, 32));

  // pass 2: exponentiate
#pragma unroll
  for (int k = 0; k < 32; ++k) x[k] = expf(x[k] - m);

  // matrix-engine reduction: row sums of the 16x64 exp block
  v2f ones; ones.x = 1.0f; ones.y = 1.0f;
  v8f acc = {0.0f, 0.0f, 0.0f, 0.0f, 0.0f, 0.0f, 0.0f, 0.0f};
#pragma unroll
  for (int c = 0; c < 16; ++c) {
    v2f a; a.x = x[2 * c]; a.y = x[2 * c + 1];
    acc = __builtin_amdgcn_wmma_f32_16x16x4_f32(
        /*neg_a=*/false, a, /*neg_b=*/false, ones,
        /*c_mod=*/(short)0, acc, /*reuse_a=*/false, /*reuse_b=*/false);
  }

  // D[m][0]: m<8 -> lane 0 vgpr m ; m>=8 -> lane 16 vgpr m-8. Route via LDS.
#pragma unroll
  for (int r = 0; r < 8; ++r) red[wv * 256 + lane * 8 + r] = acc[r];
  __syncthreads();
  float ssum = red[wv * 256 + ((eloc < 8) ? eloc : (128 + eloc - 8))];

  // pass 3: write normalized beliefs in place
#pragma unroll
  for (int c = 0; c < 16; ++c) {
    p[c * 4] = x[2 * c] / ssum;
    p[c * 4 + 1] = x[2 * c + 1] / ssum;
  }
}

// ---------------------------------------------------------------------------
extern "C" void kernel_launch(void* const* d_in, const int* in_sizes, int n_in,
                              void* d_out, int out_size, void* d_ws, size_t ws_size,
                              hipStream_t stream) {
  const float* unary = (const float*)d_in[0];            // [V,8]
  const float* pairwise = (const float*)d_in[1];         // [E,8,8]
  const float* init_msgs = (const float*)d_in[2];        // [2,E,8]
  const int* eidx = (const int*)d_in[3];                 // [2,E]
  const unsigned char* vmask = (const unsigned char*)d_in[4];  // [V,8] bool
  const unsigned char* fmask = (const unsigned char*)d_in[5];  // [E,8,8] bool

  float* vb = (float*)d_ws;                    // V*8
  float* msA = vb + NVAR * NS;                 // 2*E*8
  float* msB = msA + 2 * NEDGE * NS;           // 2*E*8

  float* out_vb = (float*)d_out;               // V*8
  float* out_fb = out_vb + NVAR * NS;          // E*64

  const float* cur = init_msgs;
  float* nxt = msA;

  for (int t = 0; t < TOTAL_STEPS; ++t) {
    init_vb_kernel<<<(NVAR * NS + 255) / 256, 256, 0, stream>>>(unary, vb);
    scatter_kernel<<<(2 * NEDGE * NS) / 256, 256, 0, stream>>>(cur, eidx, vb);
    edge_kernel<<<NEDGE / EPB, EPB, 0, stream>>>(
        cur, nxt, vb, pairwise, eidx,
        (t == TOTAL_STEPS - 1) ? out_fb : (float*)nullptr, fmask);
    cur = nxt;
    nxt = (nxt == msA) ? msB : msA;
  }

  // final variable beliefs use the last (7th-update) messages
  init_vb_kernel<<<(NVAR * NS + 255) / 256, 256, 0, stream>>>(unary, vb);
  scatter_kernel<<<(2 * NEDGE * NS) / 256, 256, 0, stream>>>(cur, eidx, vb);

  calibrate_vb_kernel<<<(NVAR + 255) / 256, 256, 0, stream>>>(vb, vmask, out_vb);
  calibrate_fb_kernel<<<NEDGE / 64, 128, 0, stream>>>(out_fb);
}